// Height1Net_28174985462431
// MI455X (gfx1250) — compile-verified
//
#include <hip/hip_runtime.h>

typedef float v4f __attribute__((ext_vector_type(4)));

#define TPB 256
#define NBUF 4            // LDS ring buffers (4 KB each)
#define LOOKAHEAD 3       // async stages in flight beyond the consumed one

// ---- gfx1250 helpers -------------------------------------------------------

// Raw 32-bit LDS byte offset of a __shared__ object (addrspacecast -> ptrtoint).
__device__ __forceinline__ unsigned lds_off_of(const void* p) {
  return (unsigned)(unsigned long long)(__attribute__((address_space(3))) const void*)p;
}

// Async 16B copy global -> LDS, ASYNCcnt-tracked; NT hint: stream-once input,
// don't let it occupy L2 (input+output streams exceed the 192MB L2).
__device__ __forceinline__ void async_copy16(unsigned lds_byte_off, const v4f* gptr) {
  asm volatile("global_load_async_to_lds_b128 %0, %1, off th:TH_LOAD_NT"
               :: "v"(lds_byte_off), "v"(gptr)
               : "memory");
}

#define S_WAIT_ASYNCCNT(n) asm volatile("s_wait_asynccnt " #n ::: "memory")

// learned_act(t) = relu(t-0.2)+relu(t-0.1)-relu(t)
//               = med3(-t, -0.1, 0) + relu(t-0.2)   (exact, piecewise-linear)
__device__ __forceinline__ float act_f(float t) {
  return __builtin_amdgcn_fmed3f(-t, -0.1f, 0.0f) + fmaxf(t - 0.2f, 0.0f);
}

// ---- kernel ----------------------------------------------------------------

extern "C" __global__ void __launch_bounds__(TPB)
h1net_stream(const float* __restrict__ xp,
             const float* __restrict__ w_in,  const float* __restrict__ b_in,
             const float* __restrict__ w_mid, const float* __restrict__ b_mid,
             const float* __restrict__ w_out, const float* __restrict__ b_out,
             float* __restrict__ yp, int n)
{
  __shared__ v4f tile[NBUF][TPB];

  // Tiny weight set -> registers (uniform kernarg-relative loads -> s_load).
  const float wi0 = w_in[0],  wi1 = w_in[1],  wi2 = w_in[2];
  const float bi0 = b_in[0],  bi1 = b_in[1],  bi2 = b_in[2];
  const float m00 = w_mid[0], m01 = w_mid[1], m02 = w_mid[2];
  const float m10 = w_mid[3], m11 = w_mid[4], m12 = w_mid[5];
  const float m20 = w_mid[6], m21 = w_mid[7], m22 = w_mid[8];
  const float bm0 = b_mid[0], bm1 = b_mid[1], bm2 = b_mid[2];
  const float wo0 = w_out[0], wo1 = w_out[1], wo2 = w_out[2];
  const float bo  = b_out[0];

  auto evalx = [&](float x) -> float {
    float a0 = act_f(fmaf(x, wi0, bi0));
    float a1 = act_f(fmaf(x, wi1, bi1));
    float a2 = act_f(fmaf(x, wi2, bi2));
    float g0 = fmaf(m02, a2, fmaf(m01, a1, fmaf(m00, a0, bm0)));
    float g1 = fmaf(m12, a2, fmaf(m11, a1, fmaf(m10, a0, bm1)));
    float g2 = fmaf(m22, a2, fmaf(m21, a1, fmaf(m20, a0, bm2)));
    float c0 = act_f(g0);
    float c1 = act_f(g1);
    float c2 = act_f(g2);
    return fmaf(wo2, c2, fmaf(wo1, c1, fmaf(wo0, c0, bo)));
  };

  const int tid    = threadIdx.x;
  const int n4     = n >> 2;          // # of float4 (fits int: n <= 2^31-1)
  const int nt     = n4 / TPB;        // # of full 256-float4 tiles
  const int stride = (int)gridDim.x;

  const v4f* xv = (const v4f*)xp;
  v4f*       yv = (v4f*)yp;

  const unsigned lbase = lds_off_of(&tile[0][tid]);  // buf b at lbase + (b<<12)

  // ---- main path: 4-deep async-to-LDS ring, lookahead 3 --------------------
  int t = (int)blockIdx.x;
  if (t              < nt) async_copy16(lbase,              xv + (t             ) * TPB + tid);
  if (t +     stride < nt) async_copy16(lbase + (1u << 12), xv + (t +     stride) * TPB + tid);
  if (t + 2 * stride < nt) async_copy16(lbase + (2u << 12), xv + (t + 2 * stride) * TPB + tid);

  int i = 0;
  for (; t < nt; t += stride, ++i) {
    const int tn = t + LOOKAHEAD * stride;
    if (tn < nt) {
      async_copy16(lbase + ((unsigned)((i + LOOKAHEAD) & (NBUF - 1)) << 12),
                   xv + tn * TPB + tid);
      // loads complete in order: <=3 outstanding => stage being consumed landed
      S_WAIT_ASYNCCNT(3);
    } else {
      S_WAIT_ASYNCCNT(0);
    }
    v4f v = tile[i & (NBUF - 1)][tid];           // ds_load_b128 (own lane's bytes)
    v4f r;
    r.x = evalx(v.x);
    r.y = evalx(v.y);
    r.z = evalx(v.z);
    r.w = evalx(v.w);
    __builtin_nontemporal_store(r, &yv[t * TPB + tid]);  // write-once stream (NT)
  }

  // ---- tails (empty for 16384x2048, kept for safety) -----------------------
  const int gtid    = (int)blockIdx.x * TPB + tid;
  const int gstride = (int)gridDim.x * TPB;

  for (int idx = nt * TPB + gtid; idx < n4; idx += gstride) {
    v4f v = __builtin_nontemporal_load(&xv[idx]);
    v4f r;
    r.x = evalx(v.x); r.y = evalx(v.y); r.z = evalx(v.z); r.w = evalx(v.w);
    __builtin_nontemporal_store(r, &yv[idx]);
  }
  for (int idx = (n4 << 2) + gtid; idx < n; idx += gstride) {
    yp[idx] = evalx(xp[idx]);
  }
}

// ---- launch ----------------------------------------------------------------

extern "C" void kernel_launch(void* const* d_in, const int* in_sizes, int n_in,
                              void* d_out, int out_size, void* d_ws, size_t ws_size,
                              hipStream_t stream) {
  const float* x     = (const float*)d_in[0];
  const float* w_in  = (const float*)d_in[1];
  const float* b_in  = (const float*)d_in[2];
  const float* w_mid = (const float*)d_in[3];
  const float* b_mid = (const float*)d_in[4];
  const float* w_out = (const float*)d_in[5];
  const float* b_out = (const float*)d_in[6];
  float*       y     = (float*)d_out;

  const int n  = out_size;                 // 16384*2048 elements
  const int nt = (n >> 2) / TPB;           // full tiles
  int grid = nt > 0 ? nt : 1;
  if (grid > 4096) grid = 4096;            // grid-stride, ~8 tiles/block

  h1net_stream<<<grid, TPB, 0, stream>>>(x, w_in, b_in, w_mid, b_mid,
                                         w_out, b_out, y, n);
}